// NGCF_54022098649341
// MI455X (gfx1250) — compile-verified
//
#include <hip/hip_runtime.h>
#include <hip/hip_bf16.h>

#if defined(__HIP_DEVICE_COMPILE__)
#if !__has_builtin(__builtin_amdgcn_wmma_f32_16x16x4_f32)
#error "missing __builtin_amdgcn_wmma_f32_16x16x4_f32 on this toolchain (device pass)"
#endif
#endif

#define N_USER 60000
#define N_ITEM 40000
#define N_TOT  100000
#define EMB    128
#define NNZ    1600000

typedef __attribute__((ext_vector_type(2))) float v2f;
typedef __attribute__((ext_vector_type(8))) float v8f;

__device__ __forceinline__ float leaky(float x, float s) {
    return x >= 0.0f ? x : s * x;
}

// ---------------------------------------------------------------------------
// K1: tiny user-feature MLP: feats(384) -> 64 -> 128.  One block, 128 threads.
// ---------------------------------------------------------------------------
__global__ void mlp_kernel(const int* age, const int* date, const int* sex,
                           const float* age_tab, const float* date_tab, const float* sex_tab,
                           const float* lin1_w, const float* lin1_b,
                           const float* lin2_w, const float* lin2_b,
                           float* mlp_out) {
    __shared__ float feats[3 * EMB];
    __shared__ float h[EMB / 2];
    int tid = threadIdx.x;
    int a = age[0], d = date[0], s = sex[0];
    feats[tid]           = age_tab[a * EMB + tid];
    feats[EMB + tid]     = date_tab[d * EMB + tid];
    feats[2 * EMB + tid] = sex_tab[s * EMB + tid];
    __syncthreads();
    if (tid < EMB / 2) {
        float acc = lin1_b[tid];
        const float* wrow = lin1_w + tid * (3 * EMB);
        for (int i = 0; i < 3 * EMB; ++i) acc += feats[i] * wrow[i];
        h[tid] = leaky(acc, 0.01f);
    }
    __syncthreads();
    float acc = lin2_b[tid];
    const float* wrow = lin2_w + tid * (EMB / 2);
    for (int i = 0; i < EMB / 2; ++i) acc += h[i] * wrow[i];
    mlp_out[tid] = leaky(acc, 0.01f);
}

// ---------------------------------------------------------------------------
// K2: build E0 = concat(user_emb (row u_id[0] blended 50/50 with mlp), item_emb)
// float4 granularity: N_TOT*32 threads.
// ---------------------------------------------------------------------------
__global__ void init_e_kernel(const float* user_emb, const float* item_emb,
                              const int* u_id, const float* mlp_out, float* E) {
    long long idx = (long long)blockIdx.x * blockDim.x + threadIdx.x;
    if (idx >= (long long)N_TOT * (EMB / 4)) return;
    int row = (int)(idx >> 5);
    int q   = (int)(idx & 31);
    float4 v;
    if (row < N_USER) {
        v = ((const float4*)user_emb)[(long long)row * 32 + q];
        if (row == u_id[0]) {
            float4 m = ((const float4*)mlp_out)[q];
            v.x = 0.5f * (v.x + m.x);
            v.y = 0.5f * (v.y + m.y);
            v.z = 0.5f * (v.z + m.z);
            v.w = 0.5f * (v.w + m.w);
        }
    } else {
        v = ((const float4*)item_emb)[(long long)(row - N_USER) * 32 + q];
    }
    ((float4*)E)[idx] = v;
}

// ---------------------------------------------------------------------------
// K3: zero LE (float4 granularity).
// ---------------------------------------------------------------------------
__global__ void zero_kernel(float4* p, long long n4) {
    long long idx = (long long)blockIdx.x * blockDim.x + threadIdx.x;
    if (idx < n4) p[idx] = make_float4(0.f, 0.f, 0.f, 0.f);
}

// ---------------------------------------------------------------------------
// K4: SpMM scatter:  LE[rows[nz]] += vals[nz] * E[cols[nz]]
// One wave (32 lanes) per nonzero; each lane owns 4 consecutive floats.
// fp32 atomics resolve in L2 (E and LE are L2-resident: 51 MB each vs 192 MB).
// ---------------------------------------------------------------------------
__global__ void spmm_kernel(const int* __restrict__ rows, const int* __restrict__ cols,
                            const float* __restrict__ vals,
                            const float* __restrict__ E, float* __restrict__ LE) {
    int nz = blockIdx.x * 8 + (threadIdx.x >> 5);
    if (nz >= NNZ) return;
    int lane = threadIdx.x & 31;
    int r = rows[nz];
    int c = cols[nz];
    float v = vals[nz];
    float4 e = ((const float4*)(E + (long long)c * EMB))[lane];
    float* dst = LE + (long long)r * EMB + lane * 4;
    atomicAdd(dst + 0, v * e.x);
    atomicAdd(dst + 1, v * e.y);
    atomicAdd(dst + 2, v * e.z);
    atomicAdd(dst + 3, v * e.w);
}

// ---------------------------------------------------------------------------
// K5: fused dense layer (fp32 WMMA):
//   E <- leaky( (LE+E) @ W1^T + (LE*E) @ W2^T + 2*b1 + b2 , 0.2 )   (in place)
// Block = 256 threads = 8 waves; block owns a 16-row tile; LDS-staged A1/A2
// tiles (race-free in-place update); wave w owns 16 output columns.
// A operand (16x4 f32): lane -> (m = lane&15, koff = k0 + 2*(lane>>4)), float2.
// B operand (4x16 f32): same addressing against W stored [n][k] row-major.
// ---------------------------------------------------------------------------
__global__ void gemm_kernel(float* __restrict__ E, const float* __restrict__ LE,
                            const float* __restrict__ W1, const float* __restrict__ W2,
                            const float* __restrict__ b1, const float* __restrict__ b2) {
    __shared__ float4 sa1[16 * EMB / 4];   // (LE + E) tile, 8 KB
    __shared__ float4 sa2[16 * EMB / 4];   // (LE * E) tile, 8 KB

    int tid = threadIdx.x;
    long long rb = blockIdx.x;                       // 16-row block
    const float4* Erow  = (const float4*)(E  + rb * 16 * EMB);
    const float4* LErow = (const float4*)(LE + rb * 16 * EMB);

    // stage: 512 float4 per matrix across 256 threads
    for (int i = tid; i < 16 * EMB / 4; i += 256) {
        float4 e = Erow[i];
        float4 l = LErow[i];
        sa1[i] = make_float4(l.x + e.x, l.y + e.y, l.z + e.z, l.w + e.w);
        sa2[i] = make_float4(l.x * e.x, l.y * e.y, l.z * e.z, l.w * e.w);
    }
    __syncthreads();

    const float* A1 = (const float*)sa1;
    const float* A2 = (const float*)sa2;

    int wave   = tid >> 5;        // output n-tile (0..7)
    int lane   = tid & 31;
    int lane15 = lane & 15;
    int hi     = lane >> 4;       // 0 or 1
    int koffb  = hi << 1;         // +0 or +2 within 4-wide k step
    int n      = (wave << 4) + lane15;   // global output column for B operand

    v8f acc = {0.f, 0.f, 0.f, 0.f, 0.f, 0.f, 0.f, 0.f};

    const float* w1n = W1 + (long long)n * EMB;
    const float* w2n = W2 + (long long)n * EMB;
    const float* a1m = A1 + (lane15 << 7);   // row m = lane&15 of staged tile
    const float* a2m = A2 + (lane15 << 7);

#if defined(__HIP_DEVICE_COMPILE__)
#pragma unroll 4
    for (int kb = 0; kb < EMB / 4; ++kb) {
        int koff = (kb << 2) + koffb;
        v2f a1 = *(const v2f*)(a1m + koff);
        v2f a2 = *(const v2f*)(a2m + koff);
        v2f bb1 = *(const v2f*)(w1n + koff);
        v2f bb2 = *(const v2f*)(w2n + koff);
        acc = __builtin_amdgcn_wmma_f32_16x16x4_f32(false, a1, false, bb1,
                                                    (short)0, acc, false, false);
        acc = __builtin_amdgcn_wmma_f32_16x16x4_f32(false, a2, false, bb2,
                                                    (short)0, acc, false, false);
    }
#endif

    float bias = 2.0f * b1[n] + b2[n];
#pragma unroll
    for (int r = 0; r < 8; ++r) {
        int row = (int)(rb * 16) + r + (hi << 3);    // C/D: M = r + 8*(lane>=16)
        float v = acc[r] + bias;
        v = leaky(v, 0.2f);
        E[(long long)row * EMB + n] = v;
    }
}

// ---------------------------------------------------------------------------
// K6: gather + (optional) row-normalize into output column block `layer`.
// One wave per gathered row (3072 rows); lane owns a float4.
// out layout: [users 1024x512][pos 1024x512][neg 1024x512]
// ---------------------------------------------------------------------------
__global__ void gather_kernel(const float* __restrict__ E,
                              const int* __restrict__ u_id,
                              const int* __restrict__ pos_item,
                              const int* __restrict__ neg_item,
                              float* __restrict__ out, int layer, int do_norm) {
    int g = blockIdx.x * 8 + (threadIdx.x >> 5);
    if (g >= 3072) return;
    int lane = threadIdx.x & 31;
    int sec = g >> 10;
    int j   = g & 1023;
    int row;
    if (sec == 0)      row = u_id[j];
    else if (sec == 1) row = N_USER + pos_item[j];
    else               row = N_USER + neg_item[j];

    float4 v = ((const float4*)(E + (long long)row * EMB))[lane];
    float scale = 1.0f;
    if (do_norm) {
        float ss = v.x * v.x + v.y * v.y + v.z * v.z + v.w * v.w;
        for (int o = 16; o > 0; o >>= 1) ss += __shfl_xor(ss, o, 32);
        float nrm = sqrtf(ss);
        scale = 1.0f / fmaxf(nrm, 1e-12f);
    }
    float4 o4 = make_float4(v.x * scale, v.y * scale, v.z * scale, v.w * scale);
    ((float4*)(out + (long long)g * 512 + layer * EMB))[lane] = o4;
}

// ---------------------------------------------------------------------------
extern "C" void kernel_launch(void* const* d_in, const int* in_sizes, int n_in,
                              void* d_out, int out_size, void* d_ws, size_t ws_size,
                              hipStream_t stream) {
    const int*   u_id     = (const int*)  d_in[0];
    const int*   pos_item = (const int*)  d_in[1];
    const int*   neg_item = (const int*)  d_in[2];
    const int*   age      = (const int*)  d_in[3];
    const int*   date     = (const int*)  d_in[4];
    const int*   sex      = (const int*)  d_in[5];
    const int*   lap_rows = (const int*)  d_in[6];
    const int*   lap_cols = (const int*)  d_in[7];
    const float* lap_vals = (const float*)d_in[8];
    const float* user_emb = (const float*)d_in[9];
    const float* item_emb = (const float*)d_in[10];
    const float* age_tab  = (const float*)d_in[11];
    const float* date_tab = (const float*)d_in[12];
    const float* sex_tab  = (const float*)d_in[13];
    const float* lin1_w   = (const float*)d_in[14];
    const float* lin1_b   = (const float*)d_in[15];
    const float* lin2_w   = (const float*)d_in[16];
    const float* lin2_b   = (const float*)d_in[17];
    const float* w1_w     = (const float*)d_in[18];
    const float* w1_b     = (const float*)d_in[19];
    const float* w2_w     = (const float*)d_in[20];
    const float* w2_b     = (const float*)d_in[21];

    float* out = (float*)d_out;

    const long long EBYTES = (long long)N_TOT * EMB * sizeof(float);   // 51.2 MB
    char*  ws      = (char*)d_ws;
    float* mlp_out = (float*)ws;                       // 512 B
    float* E       = (float*)(ws + 1024);              // N x 128
    float* LE      = (float*)(ws + 1024 + EBYTES);     // N x 128

    mlp_kernel<<<1, 128, 0, stream>>>(age, date, sex, age_tab, date_tab, sex_tab,
                                      lin1_w, lin1_b, lin2_w, lin2_b, mlp_out);

    init_e_kernel<<<(N_TOT * 32 + 255) / 256, 256, 0, stream>>>(user_emb, item_emb,
                                                                u_id, mlp_out, E);

    // layer 0 (raw E0, no normalization)
    gather_kernel<<<384, 256, 0, stream>>>(E, u_id, pos_item, neg_item, out, 0, 0);

    for (int k = 0; k < 3; ++k) {
        zero_kernel<<<(N_TOT * 32 + 255) / 256, 256, 0, stream>>>((float4*)LE,
                                                                  (long long)N_TOT * 32);
        spmm_kernel<<<NNZ / 8, 256, 0, stream>>>(lap_rows, lap_cols, lap_vals, E, LE);
        gemm_kernel<<<N_TOT / 16, 256, 0, stream>>>(E, LE,
                                                    w1_w + (long long)k * EMB * EMB,
                                                    w2_w + (long long)k * EMB * EMB,
                                                    w1_b + k * EMB,
                                                    w2_b + k * EMB);
        gather_kernel<<<384, 256, 0, stream>>>(E, u_id, pos_item, neg_item, out, k + 1, 1);
    }
}